// IntSoftmax_Piecewise_30820685316524
// MI455X (gfx1250) — compile-verified
//
#include <hip/hip_runtime.h>
#include <stdint.h>

// IntSoftmax piecewise (IBERT-style) for MI455X / gfx1250.
// Forward output == softmax_int * 2^-7 (straight-through float-softmax terms
// cancel bitwise), so only the integer path is computed.
//
// CDNA5 data path: per-wave async DMA of each 1024-float row into LDS
// (global_load_async_to_lds_b128 + s_wait_asynccnt), three LDS passes
// (quantize+max / piecewise int-exp / normalize+store), wave32 shuffle
// reductions, non-temporal b128 output stores.

#define ROW_LEN          1024
#define WAVES_PER_BLOCK  8
#define BLOCK_THREADS    (WAVES_PER_BLOCK * 32)
#define VEC_PER_LANE     8       // 8 x float4 = 32 floats per lane = 1024 / 32
#define NSEG             16

typedef float v4f __attribute__((ext_vector_type(4)));   // native vector for NT store

__device__ __forceinline__ uint32_t lds_off_u32(const void* p) {
    // Flat LDS pointer: low 32 bits are the byte offset within the WG's LDS.
    return (uint32_t)(uintptr_t)p;
}

__device__ __forceinline__ float wave_max32(float v) {
#pragma unroll
    for (int off = 16; off > 0; off >>= 1)
        v = fmaxf(v, __shfl_xor(v, off, 32));
    return v;
}

__device__ __forceinline__ float wave_sum32(float v) {
#pragma unroll
    for (int off = 16; off > 0; off >>= 1)
        v += __shfl_xor(v, off, 32);
    return v;
}

__global__ __launch_bounds__(BLOCK_THREADS)
void int_softmax_piecewise_kernel(const float* __restrict__ x,
                                  const float* __restrict__ s_ptr,
                                  const float* __restrict__ lo_bounds,
                                  const float* __restrict__ coeffs,
                                  float* __restrict__ out,
                                  int rows)
{
    __shared__ __align__(16) float stage[WAVES_PER_BLOCK * ROW_LEN]; // 32 KB row staging
    __shared__ float4 ctab[NSEG];        // {c2, c1, c0, 0} per segment (Horner order)
    __shared__ float  lotab[NSEG + 1];   // lo bounds + +inf sentinel

    const int tid  = threadIdx.x;
    const int wave = tid >> 5;
    const int lane = tid & 31;
    const int row  = blockIdx.x * WAVES_PER_BLOCK + wave;

    float* srow = &stage[wave * ROW_LEN];
    const uint32_t srow_lds = lds_off_u32(srow);
    const float4* grow4 = (const float4*)(x + (size_t)row * ROW_LEN);

    // ---- CDNA5 async DMA first: start this wave's 4 KB row transfer ASAP ----
    if (row < rows) {
#pragma unroll
        for (int k = 0; k < VEC_PER_LANE; ++k) {
            const int c = lane + k * 32;                  // coalesced 512B per issue
            uint32_t dst = srow_lds + (uint32_t)c * 16u;
            const float4* src = grow4 + c;
            asm volatile("global_load_async_to_lds_b128 %0, %1, off"
                         :: "v"(dst), "v"(src)
                         : "memory");
        }
    }

    // ---- stage lookup tables into LDS (overlaps with in-flight DMA) ----
    if (tid < NSEG) {
        ctab[tid]  = make_float4(coeffs[tid * 3 + 0], coeffs[tid * 3 + 1],
                                 coeffs[tid * 3 + 2], 0.0f);
        lotab[tid] = lo_bounds[tid];
    }
    if (tid == NSEG) lotab[NSEG] = 3.0e38f;   // sentinel: never step above seg 15

    __syncthreads();   // publish ctab/lotab (async loads are per-wave tracked)

    if (row < rows) {
        asm volatile("s_wait_asynccnt 0" ::: "memory");   // this wave's row is in LDS

        const float s  = s_ptr[0];
        const float rs = 1.0f / s;

        // ---- pass 1: x_int = floor(x/s); row max; overwrite row with x_int ----
        float m = -3.0e38f;
#pragma unroll
        for (int k = 0; k < VEC_PER_LANE; ++k) {
            const int c = lane + k * 32;
            float4 v = ((const float4*)srow)[c];
            v.x = floorf(v.x * rs);
            v.y = floorf(v.y * rs);
            v.z = floorf(v.z * rs);
            v.w = floorf(v.w * rs);
            m = fmaxf(fmaxf(fmaxf(m, v.x), fmaxf(v.y, v.z)), v.w);
            ((float4*)srow)[c] = v;           // in-place: row now holds x_int
        }
        m = wave_max32(m);

        // segment guess parameters (bounds are a floored linspace -> affine + +/-1 fixup)
        const float lo0 = lotab[0];
        const float l15 = lotab[15];
        const float gscale = (l15 > lo0) ? (15.0f / (l15 - lo0)) : 0.0f;

        // ---- pass 2: piecewise int-exp, accumulate row sum, overwrite row with e ----
        float sum = 0.0f;
#pragma unroll
        for (int k = 0; k < VEC_PER_LANE; ++k) {
            const int c = lane + k * 32;
            float4 v = ((const float4*)srow)[c];
            float4 e;
            float* vp = &v.x;
            float* ep = &e.x;
#pragma unroll
            for (int j = 0; j < 4; ++j) {
                float xi = vp[j] - m;                              // <= 0, integer-valued
                int g = (int)((xi - lo0) * gscale);                // affine segment guess
                g = g < 0 ? 0 : (g > 15 ? 15 : g);
                float bl = lotab[g];
                float bh = lotab[g + 1];
                g += (xi >= bh) ? 1 : 0;                           // side="right" semantics
                g -= (xi < bl) ? 1 : 0;
                g = g < 0 ? 0 : g;                                 // clip like reference
                float4 cf = ctab[g];                               // ds_load_b128
                float r  = fmaf(fmaf(cf.x, xi, cf.y), xi, cf.z);   // Horner, deg 2
                float ev = fmaxf(r, 0.0f);
                ev = floorf(ev * 0x1p-15f);   // floor(exp_int / 2^(30-16+1)), exact scale
                ep[j] = ev;
                sum += ev;
            }
            ((float4*)srow)[c] = e;           // in-place: row now holds exp_int
        }
        sum = wave_sum32(sum);
        const float exp_sum = fmaxf(sum, 1.0f);
        const float factor  = floorf(4294967296.0f / exp_sum);     // floor(2^32 / sum)

        // ---- pass 3: integer softmax, scale to output grid, NT store ----
        v4f* orow4 = (v4f*)(out + (size_t)row * ROW_LEN);
#pragma unroll
        for (int k = 0; k < VEC_PER_LANE; ++k) {
            const int c = lane + k * 32;
            float4 e = ((const float4*)srow)[c];
            v4f o;
            o.x = floorf((e.x * factor) * 0x1p-25f) * 0.0078125f;  // /2^25, * s_out=2^-7
            o.y = floorf((e.y * factor) * 0x1p-25f) * 0.0078125f;
            o.z = floorf((e.z * factor) * 0x1p-25f) * 0.0078125f;
            o.w = floorf((e.w * factor) * 0x1p-25f) * 0.0078125f;
            __builtin_nontemporal_store(o, &orow4[c]);             // streaming b128 store
        }
    }

    // trailing scalar output: s_out
    if (blockIdx.x == 0 && tid == 0)
        out[(size_t)rows * ROW_LEN] = 0.0078125f;
}

extern "C" void kernel_launch(void* const* d_in, const int* in_sizes, int n_in,
                              void* d_out, int out_size, void* d_ws, size_t ws_size,
                              hipStream_t stream) {
    (void)n_in; (void)out_size; (void)d_ws; (void)ws_size;
    const float* x      = (const float*)d_in[0];   // [B,H,S,S] fp32
    const float* s      = (const float*)d_in[1];   // scalar scaling_factor
    const float* lo     = (const float*)d_in[2];   // [16] lo_bounds
    // d_in[3] = hi_bounds: signature parity only (segments contiguous)
    const float* cf     = (const float*)d_in[4];   // [16,3] integer coefficients
    float*       out    = (float*)d_out;

    const int n    = in_sizes[0];
    const int rows = n / ROW_LEN;
    const int grid = (rows + WAVES_PER_BLOCK - 1) / WAVES_PER_BLOCK;

    int_softmax_piecewise_kernel<<<grid, BLOCK_THREADS, 0, stream>>>(
        x, s, lo, cf, out, rows);
}